// DeformableConv1D_19353122636333
// MI455X (gfx1250) — compile-verified
//
#include <hip/hip_runtime.h>
#include <hip/hip_bf16.h>

typedef float v2f __attribute__((ext_vector_type(2)));
typedef float v4f __attribute__((ext_vector_type(4)));
typedef float v8f __attribute__((ext_vector_type(8)));

#define B_   16
#define T_   65536
#define C_   32
#define F_   32
#define K_   5
#define TP_  (T_ - K_ + 1)      // 65532 output positions per batch
#define KDIM (K_ * C_)          // 160 = GEMM K
#define NKK  (KDIM / 4)         // 40 WMMA K-steps
#define MBLK 256                // positions per block (8 waves x 2 tiles x 16)
#define NBLK_T ((TP_ + MBLK - 1) / MBLK)   // 256
#define RED_BLOCKS 1024
#define THREADS 256

// CDNA5 async global->LDS copy (ASYNCcnt tracked), per ISA §15.18 VGLOBAL op 98.
__device__ __forceinline__ void async_load_b128(unsigned lds_off, const float* gaddr) {
  asm volatile("global_load_async_to_lds_b128 %0, %1, off"
               :: "v"(lds_off), "v"(gaddr) : "memory");
}
__device__ __forceinline__ void wait_async0() {
  asm volatile("s_wait_asynccnt 0x0" ::: "memory");
}

// ---------------- Phase 1: deterministic weighted reduction of x -------------
// partial[blk] = sum over a strided slice of x[b,t,c] * A[c],  A[c]=sum_{k,f} W[k,c,f]
__global__ __launch_bounds__(THREADS)
void reduce_kernel(const float* __restrict__ x, const float* __restrict__ W,
                   float* __restrict__ part) {
  __shared__ float A[C_];
  __shared__ float red[THREADS];
  int tid = threadIdx.x;
  if (tid < C_) {
    float a = 0.f;
    for (int k = 0; k < K_; ++k)
      for (int f = 0; f < F_; ++f)
        a += W[(k * C_ + tid) * F_ + f];
    A[tid] = a;
  }
  __syncthreads();
  const size_t N4 = (size_t)B_ * T_ * C_ / 4;
  size_t i = (size_t)blockIdx.x * blockDim.x + tid;
  const size_t stride = (size_t)gridDim.x * blockDim.x;
  const v4f* x4 = (const v4f*)x;
  float acc = 0.f;
  for (; i < N4; i += stride) {
    v4f v = x4[i];
    int cb = (int)(i & 7u) * 4;            // first channel of this float4
    acc += v.x * A[cb] + v.y * A[cb + 1] + v.z * A[cb + 2] + v.w * A[cb + 3];
  }
  red[tid] = acc;
  __syncthreads();
  for (int s = THREADS / 2; s > 0; s >>= 1) {
    if (tid < s) red[tid] += red[tid + s];
    __syncthreads();
  }
  if (tid == 0) part[blockIdx.x] = red[0];
}

// ---------------- Phase 2: edge corrections + scalar s -----------------------
__global__ __launch_bounds__(THREADS)
void scalar_kernel(const float* __restrict__ x, const float* __restrict__ W,
                   const float* __restrict__ bconv, const float* __restrict__ Wdef,
                   const float* __restrict__ part, float* __restrict__ sOut) {
  __shared__ float Wsum[KDIM];           // sum over f
  __shared__ float Hw[4 * C_], Tw[4 * C_];
  __shared__ float red[THREADS];
  int tid = threadIdx.x;
  if (tid < KDIM) {
    float a = 0.f;
    for (int f = 0; f < F_; ++f) a += W[tid * F_ + f];
    Wsum[tid] = a;
  }
  __syncthreads();
  if (tid < 4 * C_) {
    int t = tid / C_, c = tid % C_;
    float h = 0.f, tl = 0.f;
    for (int k = t + 1; k < K_; ++k) h  += Wsum[k * C_ + c];  // head weight
    for (int k = 0; k <= t;  ++k)    tl += Wsum[k * C_ + c];  // tail weight
    Hw[tid] = h; Tw[tid] = tl;
  }
  __syncthreads();
  // E = sum of edge-row corrections (first/last 4 rows per batch)
  int r = tid / C_;     // 0..7
  int c = tid % C_;
  float e = 0.f;
  for (int b = 0; b < B_; ++b) {
    size_t base = (size_t)b * T_ * C_;
    if (r < 4) e += x[base + (size_t)r * C_ + c] * Hw[r * C_ + c];
    else {
      int j = r - 4;
      e += x[base + (size_t)(T_ - 4 + j) * C_ + c] * Tw[j * C_ + c];
    }
  }
  float p = 0.f;
  for (int i = tid; i < RED_BLOCKS; i += THREADS) p += part[i];
  red[tid] = p - e;
  __syncthreads();
  for (int s = THREADS / 2; s > 0; s >>= 1) {
    if (tid < s) red[tid] += red[tid + s];
    __syncthreads();
  }
  if (tid == 0) {
    float bm = 0.f;
    for (int f = 0; f < F_; ++f) bm += bconv[f];
    float mean = red[0] / ((float)B_ * (float)TP_ * (float)F_) + bm / (float)F_;
    float s = 0.f;
    for (int k = 0; k < K_; ++k) {
      float pn = mean + (float)(k - K_ / 2);
      float sc = 1.f - fabsf(pn);
      if (sc < 0.f) sc = 0.f;
      s += Wdef[k] * sc;
    }
    sOut[0] = s;
  }
}

// ---------------- Phase 3: out = s * conv1d(x, W) + bconv via WMMA f32 -------
// Implicit GEMM: Out[M=pos, N=32] = Xim2col[M, 160] x Wflat[160, 32].
// Wconv (5,32,32) row-major IS Wflat(160,32) row-major.
// Each wave computes two 16-position M-tiles so each B fragment feeds 4 WMMAs.
__global__ __launch_bounds__(THREADS)
void conv_wmma_kernel(const float* __restrict__ x, const float* __restrict__ W,
                      const float* __restrict__ bconv, const float* __restrict__ sPtr,
                      float* __restrict__ out) {
  __shared__ float xs[(MBLK + K_ - 1) * C_];   // 260 rows x 32 ch = 33.3 KB
  __shared__ float wperm[NKK * 32 * 4];        // per-lane B fragments, 20 KB

  const int tid = threadIdx.x;
  const int b   = blockIdx.x / NBLK_T;
  const int blk = blockIdx.x % NBLK_T;
  const int t0  = blk * MBLK;

  // Build permuted B fragments: for (kk, lane) store
  // [Wf[k0][n0], Wf[k0+1][n0], Wf[k0][n1], Wf[k0+1][n1]] contiguously (one b128 load later)
  for (int e = tid; e < NKK * 32; e += THREADS) {
    int kk = e >> 5, l = e & 31;
    int h = l >> 4, ln = l & 15;
    int kap0 = 4 * kk + 2 * h;                 // even K index
    const float* g = W + (size_t)kap0 * F_;
    float4 v;
    v.x = g[ln];            // (kap0 , n0)
    v.y = g[F_ + ln];       // (kap0+1, n0)
    v.z = g[ln + 16];       // (kap0 , n1)
    v.w = g[F_ + ln + 16];  // (kap0+1, n1)
    *reinterpret_cast<float4*>(&wperm[e * 4]) = v;
  }

  // Async-load x tile rows [t0, t0+260) straight into LDS (zero-fill past end)
  const int ROWS = MBLK + K_ - 1;
  for (int idx = tid; idx < ROWS * (C_ / 4); idx += THREADS) {
    int row  = idx >> 3;
    int coff = (idx & 7) * 4;
    float* ldst = &xs[row * C_ + coff];
    if (t0 + row < T_) {
      const float* g = x + ((size_t)b * T_ + t0 + row) * C_ + coff;
      async_load_b128((unsigned)(unsigned long long)ldst, g);
    } else {
      *reinterpret_cast<float4*>(ldst) = make_float4(0.f, 0.f, 0.f, 0.f);
    }
  }
  wait_async0();
  __syncthreads();

  const int wave  = tid >> 5;
  const int lane  = tid & 31;
  const int h     = lane >> 4;    // lane-half selects K pair
  const int ln    = lane & 15;    // A: row M; B/D: col N
  const int wbase = wave * 32;    // first local position of this wave's tiles

  v8f acc0a = {}, acc1a = {};     // M-tile 0: f 0..15, f 16..31
  v8f acc0b = {}, acc1b = {};     // M-tile 1
#pragma unroll
  for (int kk = 0; kk < NKK; ++kk) {
    int kap0 = 4 * kk + 2 * h;    // even -> (kap0, kap0+1) share tap k, adjacent channels
    int k = kap0 >> 5;
    int c = kap0 & 31;
    v2f a0 = *reinterpret_cast<const v2f*>(&xs[(wbase      + ln + k) * C_ + c]);
    v2f a1 = *reinterpret_cast<const v2f*>(&xs[(wbase + 16 + ln + k) * C_ + c]);
    v4f w4 = *reinterpret_cast<const v4f*>(&wperm[(kk * 32 + lane) * 4]);
    v2f b0 = { w4.x, w4.y };
    v2f b1 = { w4.z, w4.w };
    acc0a = __builtin_amdgcn_wmma_f32_16x16x4_f32(false, a0, false, b0, (short)0, acc0a, false, false);
    acc1a = __builtin_amdgcn_wmma_f32_16x16x4_f32(false, a0, false, b1, (short)0, acc1a, false, false);
    acc0b = __builtin_amdgcn_wmma_f32_16x16x4_f32(false, a1, false, b0, (short)0, acc0b, false, false);
    acc1b = __builtin_amdgcn_wmma_f32_16x16x4_f32(false, a1, false, b1, (short)0, acc1b, false, false);
  }

  const float s   = sPtr[0];
  const float bc0 = bconv[ln];
  const float bc1 = bconv[ln + 16];
  const int   p0  = t0 + wbase;
#pragma unroll
  for (int r = 0; r < 8; ++r) {
    int t = p0 + r + 8 * h;       // D layout: VGPR r -> row r (+8 for upper half)
    if (t < TP_) {
      size_t o = ((size_t)b * TP_ + t) * F_;
      out[o + ln]      = s * acc0a[r] + bc0;
      out[o + ln + 16] = s * acc1a[r] + bc1;
    }
    int t2 = t + 16;              // second M-tile
    if (t2 < TP_) {
      size_t o = ((size_t)b * TP_ + t2) * F_;
      out[o + ln]      = s * acc0b[r] + bc0;
      out[o + ln + 16] = s * acc1b[r] + bc1;
    }
  }
}

extern "C" void kernel_launch(void* const* d_in, const int* in_sizes, int n_in,
                              void* d_out, int out_size, void* d_ws, size_t ws_size,
                              hipStream_t stream) {
  const float* x     = (const float*)d_in[0];
  const float* W     = (const float*)d_in[1];
  const float* bconv = (const float*)d_in[2];
  const float* Wdef  = (const float*)d_in[3];
  float* out = (float*)d_out;
  float* ws  = (float*)d_ws;
  float* sScalar = ws;          // ws[0]           : scalar s
  float* part    = ws + 1;      // ws[1..1024]     : per-block partial sums

  reduce_kernel<<<RED_BLOCKS, THREADS, 0, stream>>>(x, W, part);
  scalar_kernel<<<1, THREADS, 0, stream>>>(x, W, bconv, Wdef, part, sScalar);
  conv_wmma_kernel<<<B_ * NBLK_T, THREADS, 0, stream>>>(x, W, bconv, sScalar, out);
}